// ContrastLoss_22557168238879
// MI455X (gfx1250) — compile-verified
//
#include <hip/hip_runtime.h>
#include <math.h>

// ---------------- problem constants (static in the reference) ----------------
#define NVID   2      // B
#define NLOC   2048   // M
#define TLEN   4096   // T
#define KSEG   16     // K segments per (b,m)  -> one WMMA M-tile, one TDM gather
#define DT     150    // delta_t
#define KPAD   152    // basis K padded to multiple of 4 (WMMA K-step = 4)
#define BIN0   4      // first band bin (freqs >= 40/60 Hz)
#define NBIN   17     // bins 4..20 inclusive (freqs <= 250/60 Hz)
#define NCOL   48     // 3 WMMA N-tiles: [0..16]=cos bins, [17..33]=sin bins, rest zero
#define WAVES  2      // waves per block (keeps static LDS < 64KB)
#define SEGFL  (KSEG * KPAD)   // floats reserved per wave (rows packed at stride DT)

typedef __attribute__((ext_vector_type(2)))  float v2f;
typedef __attribute__((ext_vector_type(8)))  float v8f;
typedef __attribute__((ext_vector_type(4)))  unsigned int u32x4;
typedef __attribute__((ext_vector_type(4)))  int i32x4;
typedef __attribute__((ext_vector_type(8)))  int i32x8;

__device__ __forceinline__ int rfl(int v) { return __builtin_amdgcn_readfirstlane(v); }

// gacc layout: [b*(NBIN+1) + j] = colsum_b[j] for j<NBIN ; [b*(NBIN+1)+NBIN] = sumSq_b
__global__ void cl_zero(float* __restrict__ g) {
    if (threadIdx.x < NVID * (NBIN + 1)) g[threadIdx.x] = 0.0f;
}

__global__ __launch_bounds__(WAVES * 32)
void cl_main(const float* __restrict__ x,      // (B,M,T) fp32
             const int*   __restrict__ offs,   // (B,M,K) int32
             float* __restrict__ gacc)
{
    __shared__ float basisT[NCOL * KPAD];            // 29184 B, transposed: [col][k]
    __shared__ float seg[WAVES * SEGFL];             // 19456 B; TDM gather target
    __shared__ float accum[NVID * (NBIN + 1)];       // block-level partials

    const int tid  = threadIdx.x;
    const int lane = tid & 31;
    const int wave = tid >> 5;

    const int task = blockIdx.x * WAVES + wave;      // (b,m) index, < NVID*NLOC
    const int b    = task / NLOC;
    float* myseg   = &seg[wave * SEGFL];
    const long xbase = (long)task * TLEN;
    const int  obase = task * KSEG;

    // ---- issue TDM gather: 16 rows, row r = x[xbase + off_r .. +150) -> LDS ----
    {
        const unsigned long long ga =
            (unsigned long long)(uintptr_t)(x + xbase);          // tile base address
        const unsigned lds = (unsigned)(uintptr_t)myseg;         // LDS byte offset

        u32x4 g0;
        g0.x = (unsigned)rfl((int)(0x80000001u));                // count=1, gather_mode=1, 16b idx
        g0.y = (unsigned)rfl((int)lds);                          // lds_addr
        g0.z = (unsigned)rfl((int)(unsigned)(ga & 0xFFFFFFFFu)); // global_addr[31:0]
        g0.w = (unsigned)rfl((int)((unsigned)((ga >> 32) & 0x1FFFFFFu) | (2u << 30))); // [56:32], type=2

        i32x8 g1;
        g1[0] = rfl((int)(2u << 16));          // workgroup_mask=0, data_size=4B
        g1[1] = rfl((int)(4096u << 16));       // tensor_dim0 = 4096 (low 16 in [31:16])
        g1[2] = rfl((int)(4096u << 16));       // tensor_dim0 hi=0 | tensor_dim1=4096 low
        g1[3] = rfl((int)(150u << 16));        // tensor_dim1 hi=0 | tile_dim0 = 150
        g1[4] = rfl((int)KSEG);                // tile_dim1 = 16 valid indices
        g1[5] = rfl(1);                        // tensor_dim0_stride = 1
        g1[6] = rfl(0);
        g1[7] = rfl(0);

        i32x4 g2, g3;                          // 16-bit row indices = segment offsets
        #pragma unroll
        for (int p = 0; p < 4; ++p) {
            const int a0 = offs[obase + 2 * p];
            const int a1 = offs[obase + 2 * p + 1];
            g2[p] = rfl((a0 & 0xFFFF) | (a1 << 16));
        }
        #pragma unroll
        for (int p = 0; p < 4; ++p) {
            const int a0 = offs[obase + 8 + 2 * p];
            const int a1 = offs[obase + 8 + 2 * p + 1];
            g3[p] = rfl((a0 & 0xFFFF) | (a1 << 16));
        }
#if defined(__clang_major__) && (__clang_major__ >= 23)
        i32x8 g4 = {};
        __builtin_amdgcn_tensor_load_to_lds(g0, g1, g2, g3, g4, 0);
#else
        __builtin_amdgcn_tensor_load_to_lds(g0, g1, g2, g3, 0);
#endif
    }

    // zero LDS past row 15 (rows packed at stride DT; last WMMA K-step overhangs by 2)
    myseg[KSEG * DT + lane] = 0.0f;

    // ---- build DFT basis (cos/sin for bins 4..20) while the TDM gather runs ----
    for (int i = tid; i < NCOL * KPAD; i += WAVES * 32) {
        const int c = i / KPAD;
        const int k = i - c * KPAD;
        float v = 0.0f;
        if (k < DT) {
            if (c < NBIN) {
                v = cosf(6.28318530717958647692f * (float)((c + BIN0) * k) / (float)DT);
            } else if (c < 2 * NBIN) {
                v = sinf(6.28318530717958647692f * (float)((c - NBIN + BIN0) * k) / (float)DT);
            }
        }
        basisT[i] = v;
    }
    for (int i = tid; i < NVID * (NBIN + 1); i += WAVES * 32) accum[i] = 0.0f;
    __syncthreads();
    __builtin_amdgcn_s_wait_tensorcnt(0);      // segments resident in LDS

    // ---- WMMA f32 16x16x4, software-pipelined (prefetch next K-step frags) ----
    v8f acc0 = {}; v8f acc1 = {}; v8f acc2 = {};
    const int row  = lane & 15;                // A row / B column within tile
    const int koff = (lane >> 4) * 2;          // K sub-offset per half-wave
    const float* arow = &myseg[row * DT];      // seg rows packed at stride DT
    const float* bp0  = &basisT[(row     ) * KPAD];
    const float* bp1  = &basisT[(row + 16) * KPAD];
    const float* bp2  = &basisT[(row + 32) * KPAD];

    int k  = koff;
    v2f a  = *(const v2f*)(arow + k);
    v2f b0 = *(const v2f*)(bp0 + k);
    v2f b1 = *(const v2f*)(bp1 + k);
    v2f b2 = *(const v2f*)(bp2 + k);
    #pragma unroll 2
    for (int step = 0; step < KPAD / 4 - 1; ++step) {
        const int kn = k + 4;
        v2f an  = *(const v2f*)(arow + kn);
        v2f b0n = *(const v2f*)(bp0 + kn);
        v2f b1n = *(const v2f*)(bp1 + kn);
        v2f b2n = *(const v2f*)(bp2 + kn);
        acc0 = __builtin_amdgcn_wmma_f32_16x16x4_f32(false, a, false, b0, (short)0, acc0, false, false);
        acc1 = __builtin_amdgcn_wmma_f32_16x16x4_f32(false, a, false, b1, (short)0, acc1, false, false);
        acc2 = __builtin_amdgcn_wmma_f32_16x16x4_f32(false, a, false, b2, (short)0, acc2, false, false);
        a = an; b0 = b0n; b1 = b1n; b2 = b2n; k = kn;
    }
    acc0 = __builtin_amdgcn_wmma_f32_16x16x4_f32(false, a, false, b0, (short)0, acc0, false, false);
    acc1 = __builtin_amdgcn_wmma_f32_16x16x4_f32(false, a, false, b1, (short)0, acc1, false, false);
    acc2 = __builtin_amdgcn_wmma_f32_16x16x4_f32(false, a, false, b2, (short)0, acc2, false, false);

    // ---- spill D tiles to LDS (reuse this wave's seg region; intra-wave ordered) ----
    float* ds = myseg;                         // need 16*48 floats <= SEGFL available
    const int rhi = (lane >> 4) * 8;
    const int c0  = lane & 15;
    #pragma unroll
    for (int v = 0; v < 8; ++v) {
        ds[(rhi + v) * NCOL + c0     ] = acc0[v];
        ds[(rhi + v) * NCOL + c0 + 16] = acc1[v];
        ds[(rhi + v) * NCOL + c0 + 32] = acc2[v];
    }

    // ---- per-row: psd = re^2+im^2, normalize, accumulate colsum & sumSq ----
    if (lane < KSEG) {
        const int r = lane;
        float psd[NBIN];
        float s = 0.0f;
        #pragma unroll
        for (int j = 0; j < NBIN; ++j) {
            const float re = ds[r * NCOL + j];
            const float im = ds[r * NCOL + NBIN + j];
            const float p  = re * re + im * im;
            psd[j] = p; s += p;
        }
        const float inv = 1.0f / s;
        float sq = 0.0f;
        #pragma unroll
        for (int j = 0; j < NBIN; ++j) {
            const float q = psd[j] * inv;
            sq += q * q;
            atomicAdd(&accum[b * (NBIN + 1) + j], q);
        }
        atomicAdd(&accum[b * (NBIN + 1) + NBIN], sq);
    }
    __syncthreads();

    // ---- one global-atomic funnel per block ----
    for (int i = tid; i < NVID * (NBIN + 1); i += WAVES * 32)
        atomicAdd(&gacc[i], accum[i]);
}

__global__ void cl_final(const float* __restrict__ g, float* __restrict__ out) {
    const double N  = (double)(NLOC * KSEG);     // 32768 rows per video
    const double Fb = (double)NBIN;
    const double sqA = (double)g[NBIN];
    const double sqB = (double)g[(NBIN + 1) + NBIN];
    double aa = 0.0, bb = 0.0, ab = 0.0;
    for (int j = 0; j < NBIN; ++j) {
        const double ca = (double)g[j];
        const double cb = (double)g[(NBIN + 1) + j];
        aa += ca * ca; bb += cb * cb; ab += ca * cb;
    }
    const double pAA = (2.0 * N * sqA - 2.0 * aa) / Fb;
    const double pBB = (2.0 * N * sqB - 2.0 * bb) / Fb;
    const double pos1 = 0.5 * (pAA + pBB) / (N * (N - 1.0));
    const double pAB = (N * sqA + N * sqB - 2.0 * ab) / Fb;
    const double pos2 = pAB / (N * N);
    out[0] = (float)(pos1 + pos2);
}

extern "C" void kernel_launch(void* const* d_in, const int* in_sizes, int n_in,
                              void* d_out, int out_size, void* d_ws, size_t ws_size,
                              hipStream_t stream) {
    const float* x    = (const float*)d_in[0];   // model_output (B,M,T) fp32
    const int*   offs = (const int*)d_in[1];     // offsets (B,M,K) int32
    // d_in[2] = delta_t (==150, static band selection mirrors the reference)
    float* gacc = (float*)d_ws;                  // 36 floats of scratch
    float* out  = (float*)d_out;

    cl_zero<<<1, 64, 0, stream>>>(gacc);
    cl_main<<<(NVID * NLOC) / WAVES, WAVES * 32, 0, stream>>>(x, offs, gacc);
    cl_final<<<1, 1, 0, stream>>>(gacc, out);
}